// mask_loss_35021163331957
// MI455X (gfx1250) — compile-verified
//
#include <hip/hip_runtime.h>
#include <stdint.h>

// Problem constants (match the reference)
#define BB 16
#define CC 256
#define HH 80
#define WWIDTH 80
#define NBOX 128
#define RADIUS 10.0f
#define WFG 0.02
#define WBG 0.01

static constexpr int HW        = HH * WWIDTH;          // 6400
static constexpr int NTOT      = BB * CC * HW;         // 26,214,400
static constexpr int NV4       = NTOT / 4;             // 6,553,600 float4s
static constexpr int MASKCELLS = BB * HW;              // 102,400
static constexpr int MASKWORDS = MASKCELLS / 32;       // 3,200 (12.8 KB)
static constexpr int PASS_BLOCKS = 2048;

typedef __attribute__((ext_vector_type(2))) float v2f;
typedef __attribute__((ext_vector_type(8))) float v8f;

// ---------------------------------------------------------------------------
// CDNA5 async global->LDS copy (ASYNCcnt-tracked, no VGPR data return).
// vdst VGPR holds the LDS byte offset (low 32 bits of a flat LDS pointer),
// vaddr pair holds the 64-bit global address, saddr = off.
// ---------------------------------------------------------------------------
__device__ __forceinline__ void async_copy_b128(void* lds_dst, const void* gsrc) {
    uint32_t lds = (uint32_t)(uintptr_t)lds_dst;   // LDS aperture: low 32 bits = LDS offset
    asm volatile("global_load_async_to_lds_b128 %0, %1, off"
                 :
                 : "v"(lds), "v"(gsrc)
                 : "memory");
}
__device__ __forceinline__ void wait_async_le(int keep2) {
    if (keep2) asm volatile("s_wait_asynccnt 2" ::: "memory");
    else       asm volatile("s_wait_asynccnt 0" ::: "memory");
}

// ---------------------------------------------------------------------------
// Dual 32-lane reduction with one V_WMMA_F32_16X16X4_F32.
//   A.x = s_bg per lane, A.y = s_fg per lane  (A is 16x4, K striped over
//   {vgpr, lane-half});  B selects bg-sums into columns 0-7 and fg-sums into
//   columns 8-15.  D[m,n<8] = s_bg[m]+s_bg[m+16], D[m,n>=8] = same for fg.
//   Per-lane sum of the 8 D VGPRs + shfl_xor(16) completes the reduction.
//   Requires EXEC == all 1s (true at the call site: uniform control flow).
// ---------------------------------------------------------------------------
__device__ __forceinline__ void wave_reduce2_wmma(float s_bg, float s_fg,
                                                  float& bg_out, float& fg_out) {
    int lane = threadIdx.x & 31;
    v2f A; A.x = s_bg; A.y = s_fg;
    float sel = ((lane & 15) < 8) ? 1.0f : 0.0f;
    v2f Bm; Bm.x = sel; Bm.y = 1.0f - sel;
    v8f Cz = {};
    v8f D = __builtin_amdgcn_wmma_f32_16x16x4_f32(
        /*neg_a=*/false, A, /*neg_b=*/false, Bm,
        /*c_mod=*/(short)0, Cz, /*reuse_a=*/false, /*reuse_b=*/false);
    float t = ((D[0] + D[1]) + (D[2] + D[3])) + ((D[4] + D[5]) + (D[6] + D[7]));
    t += __shfl_xor(t, 16);
    bg_out = __shfl(t, 0);   // columns 0-7 hold total bg
    fg_out = __shfl(t, 8);   // columns 8-15 hold total fg
}

// ---------------------------------------------------------------------------
// 0) zero accumulators + mask bits (needed every call: mask kernel writes
//    ballots non-atomically and ws is poisoned before first call)
// ---------------------------------------------------------------------------
__global__ __launch_bounds__(256) void k_init(double* acc, uint32_t* bits) {
    int t = threadIdx.x;
    if (t < 3) acc[t] = 0.0;
    for (int i = t; i < MASKWORDS; i += 256) bits[i] = 0u;
}

// ---------------------------------------------------------------------------
// 1) rasterize 128 boxes into a bit-packed [B,H,W] foreground mask.
//    Box params are lane-uniform -> scalar (SMEM) loads; one ballot per wave.
// ---------------------------------------------------------------------------
__global__ __launch_bounds__(256) void k_mask(const float* __restrict__ target,
                                              uint32_t* __restrict__ bits) {
    int ci  = blockIdx.x * 256 + threadIdx.x;      // cell index < 102400
    int b   = ci / HW;
    int rem = ci - b * HW;
    int row = rem / WWIDTH;
    int col = rem - row * WWIDTH;
    float frow = (float)row, fcol = (float)col;

    bool hit = false;
    for (int n = 0; n < NBOX; ++n) {
        const float* tg = target + 6 * n;
        int bidx = (int)tg[0];
        if (bidx != b) continue;                   // wave-uniform branch
        float cx = tg[2] * (float)WWIDTH, cy = tg[3] * (float)HH;
        float w  = tg[4] * (float)WWIDTH, h  = tg[5] * (float)HH;
        float rw = floorf(w / RADIUS); if (rw == 0.0f) rw = 1.0f;
        float rh = floorf(h / RADIUS); if (rh == 0.0f) rh = 1.0f;
        float x1c = cx - 0.5f * w, x2c = cx + 0.5f * w;
        float y1c = cy - 0.5f * h, y2c = cy + 0.5f * h;
        float x1 = (x1c - rw > 0.0f) ? floorf(x1c) - rw : 0.0f;
        float y1 = (y1c - rh > 0.0f) ? floorf(y1c) - rh : 0.0f;
        float x2 = (x2c + rw < (float)WWIDTH) ? ceilf(x2c) + rw : (float)WWIDTH;
        float y2 = (y2c + rh < (float)HH)     ? ceilf(y2c) + rh : (float)HH;
        if (frow >= y1 && frow < y2 && fcol >= x1 && fcol < x2) hit = true;
    }
    unsigned long long bal = __ballot(hit);        // wave32 -> low 32 bits
    if ((threadIdx.x & 31) == 0) bits[ci >> 5] = (uint32_t)bal;
}

// ---------------------------------------------------------------------------
// 2/3) streaming pass. mode 0: acc[0]+=SUM bg*|d|, acc[1]+=SUM fg*|d|
//      mode 1: thr = acc[0]/N; acc[2]+=SUM (bg*|d| where >= thr)
//      Double-buffered async global->LDS (b128 per lane, both tensors).
// ---------------------------------------------------------------------------
__global__ __launch_bounds__(256) void k_pass(const float4* __restrict__ fs,
                                              const float4* __restrict__ ft,
                                              const uint32_t* __restrict__ bits,
                                              double* __restrict__ acc,
                                              int mode) {
    __shared__ float4 bufS[2][256];
    __shared__ float4 bufT[2][256];

    const int tid    = threadIdx.x;
    const int stride = gridDim.x * blockDim.x;     // in float4 units
    int i = blockIdx.x * blockDim.x + tid;

    float thr = 0.0f;
    if (mode) thr = (float)(acc[0] / (double)NTOT);

    float s_bg = 0.0f, s_fg = 0.0f;
    int cur = 0;

    if (i < NV4) {
        async_copy_b128(&bufS[0][tid], fs + i);
        async_copy_b128(&bufT[0][tid], ft + i);
    }
    while (i < NV4) {                              // NV4 % 256 == 0 -> block-uniform trip count
        int nxt = i + stride;
        if (nxt < NV4) {
            async_copy_b128(&bufS[cur ^ 1][tid], fs + nxt);
            async_copy_b128(&bufT[cur ^ 1][tid], ft + nxt);
            wait_async_le(1);                      // previous pair complete
        } else {
            wait_async_le(0);
        }
        float4 sv = bufS[cur][tid];
        float4 tv = bufT[cur][tid];
        float a0 = fabsf(tv.x - sv.x), a1 = fabsf(tv.y - sv.y);
        float a2 = fabsf(tv.z - sv.z), a3 = fabsf(tv.w - sv.w);

        // index math: i flattens [B, C, H, W/4]
        int w4  = i % 20;
        int bch = i / 20;                          // (b*C + c)*H + h
        int h   = bch % HH;
        int bc  = bch / HH;
        int b   = bc >> 8;                         // / CC
        int cell = b * HW + h * WWIDTH + (w4 << 2);
        uint32_t word = bits[cell >> 5];
        int sh = cell & 31;                        // w%4==0 -> 4 bits in one word
        float m0 = (float)((word >> (sh + 0)) & 1u);
        float m1 = (float)((word >> (sh + 1)) & 1u);
        float m2 = (float)((word >> (sh + 2)) & 1u);
        float m3 = (float)((word >> (sh + 3)) & 1u);

        if (mode == 0) {
            float fg  = m0 * a0 + m1 * a1 + m2 * a2 + m3 * a3;
            float all = a0 + a1 + a2 + a3;
            s_fg += fg;
            s_bg += all - fg;
        } else {
            float v0 = (1.0f - m0) * a0;
            float v1 = (1.0f - m1) * a1;
            float v2 = (1.0f - m2) * a2;
            float v3 = (1.0f - m3) * a3;
            s_bg += (v0 >= thr ? v0 : 0.0f) + (v1 >= thr ? v1 : 0.0f)
                  + (v2 >= thr ? v2 : 0.0f) + (v3 >= thr ? v3 : 0.0f);
        }
        i = nxt;
        cur ^= 1;
    }

    // wave32 dual reduction via one WMMA (EXEC is all-1s here: uniform flow)
    float bgw, fgw;
    wave_reduce2_wmma(s_bg, s_fg, bgw, fgw);

    __shared__ float wbg[8], wfg[8];
    int wave = tid >> 5;
    if ((tid & 31) == 0) { wbg[wave] = bgw; wfg[wave] = fgw; }
    __syncthreads();
    if (tid == 0) {
        float tb = 0.0f, tf = 0.0f;
        for (int k = 0; k < 8; ++k) { tb += wbg[k]; tf += wfg[k]; }
        if (mode == 0) {
            atomicAdd(&acc[0], (double)tb);
            atomicAdd(&acc[1], (double)tf);
        } else {
            atomicAdd(&acc[2], (double)tb);
        }
    }
}

// ---------------------------------------------------------------------------
// 4) final scalar
// ---------------------------------------------------------------------------
__global__ void k_final(const double* __restrict__ acc, float* __restrict__ out) {
    out[0] = (float)((WFG * acc[1] + WBG * acc[2]) / (double)NTOT);
}

// ---------------------------------------------------------------------------
extern "C" void kernel_launch(void* const* d_in, const int* in_sizes, int n_in,
                              void* d_out, int out_size, void* d_ws, size_t ws_size,
                              hipStream_t stream) {
    const float* fs = (const float*)d_in[0];   // fea_s [16,256,80,80]
    const float* ft = (const float*)d_in[1];   // fea_t [16,256,80,80]
    const float* tg = (const float*)d_in[2];   // target [128,6]

    double*   acc  = (double*)d_ws;                       // 3 doubles
    uint32_t* bits = (uint32_t*)((char*)d_ws + 64);       // 3200 words (12.8 KB)

    k_init <<<1, 256, 0, stream>>>(acc, bits);
    k_mask <<<MASKCELLS / 256, 256, 0, stream>>>(tg, bits);
    k_pass <<<PASS_BLOCKS, 256, 0, stream>>>((const float4*)fs, (const float4*)ft, bits, acc, 0);
    k_pass <<<PASS_BLOCKS, 256, 0, stream>>>((const float4*)fs, (const float4*)ft, bits, acc, 1);
    k_final<<<1, 1, 0, stream>>>(acc, (float*)d_out);
}